// SplineNet_9990093930605
// MI455X (gfx1250) — compile-verified
//
#include <hip/hip_runtime.h>
#include <math.h>

#define N_NODES 50000
#define N_EDGES 800000
#define EPSB 1e-5f

typedef __attribute__((ext_vector_type(2))) float v2f;
typedef __attribute__((ext_vector_type(8))) float v8f;

// ---------------------------------------------------------------------------
// Node transform via V_WMMA_F32_16X16X4_F32:
//   T0 = X @ W[0],  T1 = X @ W[1],  Bse = X @ R + Bias
// One wave computes one 16x16 output tile for all three matrices (A reuse x3).
// A layout (16x4 f32): lanes 0-15 row M=l, K={k0,k0+1}; lanes 16-31 K={k0+2,k0+3}
// B layout (4x16 f32): v0 = K={k0 | k0+2}, v1 = K={k0+1 | k0+3}, col = lane%16
// D layout: vgpr v -> row M = v + 8*(lane>=16), col = lane%16
// ---------------------------------------------------------------------------
template<int CIN, int COUT, int OSTRIDE>
__global__ void transform_wmma(const float* __restrict__ X,
                               const float* __restrict__ W,    // 2 x CIN x COUT
                               const float* __restrict__ R,    // CIN x COUT
                               const float* __restrict__ Bias, // COUT
                               float* __restrict__ T0,
                               float* __restrict__ T1,
                               float* __restrict__ Bse,
                               int mTiles)
{
    constexpr int NT = (COUT + 15) / 16;
    const int wave  = blockIdx.x * (blockDim.x >> 5) + (threadIdx.x >> 5);
    if (wave >= mTiles * NT) return;
    const int mt   = wave / NT;
    const int nt   = wave % NT;
    const int lane = threadIdx.x & 31;
    const int half = lane >> 4;
    const int l16  = lane & 15;
    const int row  = mt * 16 + l16;
    const int col  = nt * 16 + l16;
    const bool colOK = (COUT % 16 == 0) || (col < COUT);

    const float* __restrict__ W0 = W;
    const float* __restrict__ W1 = W + CIN * COUT;

    v8f acc0 = {}; v8f acc1 = {}; v8f accR = {};

    for (int k0 = 0; k0 < CIN; k0 += 4) {
        const int ka = k0 + 2 * half;
        v2f a;
        a.x = X[row * CIN + ka];
        a.y = X[row * CIN + ka + 1];

        v2f b0 = {}; v2f b1 = {}; v2f br = {};
        if (colOK) {
            b0.x = W0[(ka + 0) * COUT + col];
            b0.y = W0[(ka + 1) * COUT + col];
            b1.x = W1[(ka + 0) * COUT + col];
            b1.y = W1[(ka + 1) * COUT + col];
            br.x = R [(ka + 0) * COUT + col];
            br.y = R [(ka + 1) * COUT + col];
        }
        acc0 = __builtin_amdgcn_wmma_f32_16x16x4_f32(false, a, false, b0, (short)0, acc0, false, false);
        acc1 = __builtin_amdgcn_wmma_f32_16x16x4_f32(false, a, false, b1, (short)0, acc1, false, false);
        accR = __builtin_amdgcn_wmma_f32_16x16x4_f32(false, a, false, br, (short)0, accR, false, false);
    }

    const float bias = colOK ? Bias[col] : 0.0f;
    if (colOK) {
        #pragma unroll
        for (int v = 0; v < 8; ++v) {
            const int r = mt * 16 + v + 8 * half;
            T0 [(size_t)r * OSTRIDE + col] = acc0[v];
            T1 [(size_t)r * OSTRIDE + col] = acc1[v];
            Bse[(size_t)r * OSTRIDE + col] = accR[v] + bias;
        }
    }
}

// ---------------------------------------------------------------------------
__global__ void degree_kernel(const int* __restrict__ dst, float* __restrict__ deg)
{
    int e = blockIdx.x * blockDim.x + threadIdx.x;
    if (e >= N_EDGES) return;
    atomicAdd(&deg[dst[e]], 1.0f);
}

// H=64 edge scatter: 16 threads/edge, float4 per thread.
__global__ void scatter64(const int*   __restrict__ src,
                          const int*   __restrict__ dst,
                          const float* __restrict__ u,
                          const float* __restrict__ T0,
                          const float* __restrict__ T1,
                          float*       __restrict__ agg)
{
    int t = blockIdx.x * blockDim.x + threadIdx.x;
    int e  = t >> 4;
    if (e >= N_EDGES) return;
    int f4 = (t & 15) << 2;
    int s = src[e], d = dst[e];
    float uu = u[e];
    float w0 = 1.0f - uu;
    const float4 a = *(const float4*)(T0 + (size_t)s * 64 + f4);
    const float4 b = *(const float4*)(T1 + (size_t)s * 64 + f4);
    float* o = agg + (size_t)d * 64 + f4;
    atomicAdd(o + 0, w0 * a.x + uu * b.x);
    atomicAdd(o + 1, w0 * a.y + uu * b.y);
    atomicAdd(o + 2, w0 * a.z + uu * b.z);
    atomicAdd(o + 3, w0 * a.w + uu * b.w);
}

// H=7 (stride 16) edge scatter: 8 threads/edge, scalar.
__global__ void scatter7(const int*   __restrict__ src,
                         const int*   __restrict__ dst,
                         const float* __restrict__ u,
                         const float* __restrict__ T0,
                         const float* __restrict__ T1,
                         float*       __restrict__ agg)
{
    int t = blockIdx.x * blockDim.x + threadIdx.x;
    int e = t >> 3;
    int c = t & 7;
    if (e >= N_EDGES || c >= 7) return;
    int s = src[e], d = dst[e];
    float uu = u[e];
    float m = (1.0f - uu) * T0[(size_t)s * 16 + c] + uu * T1[(size_t)s * 16 + c];
    atomicAdd(&agg[(size_t)d * 16 + c], m);
}

// out = ELU( BN( agg/max(deg,1) + base ) ), H = 64
__global__ void finalize_bn_elu(const float* __restrict__ agg,
                                const float* __restrict__ base,
                                const float* __restrict__ deg,
                                const float* __restrict__ G,
                                const float* __restrict__ BE,
                                const float* __restrict__ RM,
                                const float* __restrict__ RV,
                                float* __restrict__ out)
{
    int t = blockIdx.x * blockDim.x + threadIdx.x;
    if (t >= N_NODES * 64) return;
    int h = t & 63;
    int n = t >> 6;
    float d = fmaxf(deg[n], 1.0f);
    float v = agg[t] / d + base[t];
    v = (v - RM[h]) * rsqrtf(RV[h] + EPSB) * G[h] + BE[h];
    out[t] = v > 0.0f ? v : (expf(v) - 1.0f);
}

// last layer: h = agg/deg + base (7 wide, stride 16), then log_softmax
__global__ void finalize_logsoftmax(const float* __restrict__ agg,
                                    const float* __restrict__ base,
                                    const float* __restrict__ deg,
                                    float* __restrict__ out)
{
    int n = blockIdx.x * blockDim.x + threadIdx.x;
    if (n >= N_NODES) return;
    float d = fmaxf(deg[n], 1.0f);
    float v[7];
    float mx = -3.0e38f;
    #pragma unroll
    for (int c = 0; c < 7; ++c) {
        v[c] = agg[(size_t)n * 16 + c] / d + base[(size_t)n * 16 + c];
        mx = fmaxf(mx, v[c]);
    }
    float s = 0.0f;
    #pragma unroll
    for (int c = 0; c < 7; ++c) s += expf(v[c] - mx);
    float lse = mx + logf(s);
    #pragma unroll
    for (int c = 0; c < 7; ++c) out[(size_t)n * 7 + c] = v[c] - lse;
}

// ---------------------------------------------------------------------------
extern "C" void kernel_launch(void* const* d_in, const int* in_sizes, int n_in,
                              void* d_out, int out_size, void* d_ws, size_t ws_size,
                              hipStream_t stream)
{
    const float* x   = (const float*)d_in[0];
    const int*   ei  = (const int*)  d_in[1];
    const float* ea  = (const float*)d_in[2];
    const float* W0  = (const float*)d_in[3];
    const float* R0  = (const float*)d_in[4];
    const float* B0  = (const float*)d_in[5];
    const float* W1  = (const float*)d_in[6];
    const float* R1  = (const float*)d_in[7];
    const float* B1  = (const float*)d_in[8];
    const float* W2  = (const float*)d_in[9];
    const float* R2  = (const float*)d_in[10];
    const float* B2  = (const float*)d_in[11];
    const float* G0  = (const float*)d_in[12];
    const float* BE0 = (const float*)d_in[13];
    const float* RM0 = (const float*)d_in[14];
    const float* RV0 = (const float*)d_in[15];
    const float* G1  = (const float*)d_in[16];
    const float* BE1 = (const float*)d_in[17];
    const float* RM1 = (const float*)d_in[18];
    const float* RV1 = (const float*)d_in[19];

    const int* src = ei;
    const int* dst = ei + N_EDGES;

    float* ws  = (float*)d_ws;
    float* deg = ws;                       // N
    float* T0  = deg + N_NODES;            // N*64
    float* T1  = T0  + (size_t)N_NODES*64; // N*64
    float* Bse = T1  + (size_t)N_NODES*64; // N*64
    float* agg = Bse + (size_t)N_NODES*64; // N*64
    float* h   = agg + (size_t)N_NODES*64; // N*64

    const int MT = N_NODES / 16;           // 3125, exact

    // degree (same edge set for all layers)
    hipMemsetAsync(deg, 0, N_NODES * sizeof(float), stream);
    degree_kernel<<<(N_EDGES + 255) / 256, 256, 0, stream>>>(dst, deg);

    // ---- layer 0: 32 -> 64 ----
    transform_wmma<32, 64, 64><<<(MT * 4 + 7) / 8, 256, 0, stream>>>(x, W0, R0, B0, T0, T1, Bse, MT);
    hipMemsetAsync(agg, 0, (size_t)N_NODES * 64 * sizeof(float), stream);
    scatter64<<<(N_EDGES * 16) / 256, 256, 0, stream>>>(src, dst, ea, T0, T1, agg);
    finalize_bn_elu<<<(N_NODES * 64) / 256, 256, 0, stream>>>(agg, Bse, deg, G0, BE0, RM0, RV0, h);

    // ---- layer 1: 64 -> 64 ----
    transform_wmma<64, 64, 64><<<(MT * 4 + 7) / 8, 256, 0, stream>>>(h, W1, R1, B1, T0, T1, Bse, MT);
    hipMemsetAsync(agg, 0, (size_t)N_NODES * 64 * sizeof(float), stream);
    scatter64<<<(N_EDGES * 16) / 256, 256, 0, stream>>>(src, dst, ea, T0, T1, agg);
    finalize_bn_elu<<<(N_NODES * 64) / 256, 256, 0, stream>>>(agg, Bse, deg, G1, BE1, RM1, RV1, h);

    // ---- layer 2: 64 -> 7 (padded to 16-wide tiles, stride 16) ----
    transform_wmma<64, 7, 16><<<(MT + 7) / 8, 256, 0, stream>>>(h, W2, R2, B2, T0, T1, Bse, MT);
    hipMemsetAsync(agg, 0, (size_t)N_NODES * 16 * sizeof(float), stream);
    scatter7<<<(N_EDGES * 8) / 256, 256, 0, stream>>>(src, dst, ea, T0, T1, agg);
    finalize_logsoftmax<<<(N_NODES + 255) / 256, 256, 0, stream>>>(agg, Bse, deg, (float*)d_out);
}